// IEMC_nonphy_30906584662136
// MI455X (gfx1250) — compile-verified
//
#include <hip/hip_runtime.h>

typedef __attribute__((ext_vector_type(2))) float v2f;
typedef __attribute__((ext_vector_type(8))) float v8f;

#define BROWS 8192
#define DDIM  1024
#define NCLS  7
#define WAVES_PER_BLOCK 8
#define NTILES (BROWS / 16)                       // 512 tiles of 16 rows
#define MAIN_BLOCKS (NTILES / WAVES_PER_BLOCK)    // 64 blocks

// ---------------- stage 0: init workspace ----------------
__global__ void k_init(int* wsI) {
    const int t = threadIdx.x;
    if (t < 8) wsI[t] = 0;            // counts[0..6], class a
    if (t == 8) wsI[8] = 0x7FFFFFFF;  // anchor index (atomicMin target)
}

// ---------------- stage 1: bincount labels ----------------
__global__ void k_count(const int* __restrict__ label, int* wsI) {
    const int i = blockIdx.x * blockDim.x + threadIdx.x;
    if (i < BROWS) atomicAdd(&wsI[label[i]], 1);
}

// ---------------- stage 2: first class with count > 1 ----------------
__global__ void k_pick(int* wsI) {
    int a = 0;
    for (int c = NCLS - 1; c >= 0; --c)
        if (wsI[c] > 1) a = c;        // ends at smallest qualifying class (0 if none)
    wsI[7] = a;
}

// ---------------- stage 3: first occurrence of that class ----------------
__global__ void k_anchor(const int* __restrict__ label, int* wsI) {
    const int i = blockIdx.x * blockDim.x + threadIdx.x;
    if (i < BROWS && label[i] == wsI[7]) atomicMin(&wsI[8], i);
}

// ---------------- stage 4: WMMA GEMV tiles + norms + masked partial sums ----
__global__ __launch_bounds__(256) void k_main(const float* __restrict__ s_t,
                                              const float* __restrict__ s_a,
                                              const int*   __restrict__ label,
                                              const int*   __restrict__ wsI,
                                              float*       __restrict__ partials) {
    __shared__ float sd1[WAVES_PER_BLOCK][16];
    __shared__ float sd2[WAVES_PER_BLOCK][16];
    __shared__ float sblk[WAVES_PER_BLOCK][4];

    const int lane = threadIdx.x & 31;
    const int wave = threadIdx.x >> 5;
    const int tile = blockIdx.x * WAVES_PER_BLOCK + wave;
    const int r0   = tile * 16;

    int anchor = wsI[8];
    if (anchor > BROWS - 1) anchor = 0;   // safety clamp
    const int labA = wsI[7];

    // A-fragment layout for V_WMMA_F32_16X16X4_F32 (16x4, f32, wave32):
    //   lanes 0-15 : row M = lane,    VGPR0=K+0, VGPR1=K+1
    //   lanes 16-31: row M = lane-16, VGPR0=K+2, VGPR1=K+3
    const int row  = r0 + (lane & 15);
    const int koff = (lane >> 4) * 2;
    const float* pa = s_a + (size_t)row    * DDIM + koff; // rows for score1 dots
    const float* pt = s_t + (size_t)row    * DDIM + koff; // rows for score2 dots
    const float* bt = s_t + (size_t)anchor * DDIM + koff; // anchor_t -> B for s_a dots
    const float* ba = s_a + (size_t)anchor * DDIM + koff; // anchor_a -> B for s_t dots
    const bool bsel = (lane & 15) == 0;                   // B column 0 only
    const v2f zz = {0.f, 0.f};

    v8f acc1 = {};   // D col0 = s_a-rows . anchor_t
    v8f acc2 = {};   // D col0 = s_t-rows . anchor_a
    float na = 0.f, nt = 0.f;

#pragma unroll 4
    for (int k = 0; k < DDIM; k += 4) {
        v2f aa  = *(const v2f*)(pa + k);
        v2f at  = *(const v2f*)(pt + k);
        v2f b1v = *(const v2f*)(bt + k);
        v2f b2v = *(const v2f*)(ba + k);
        v2f b1 = bsel ? b1v : zz;
        v2f b2 = bsel ? b2v : zz;
        acc1 = __builtin_amdgcn_wmma_f32_16x16x4_f32(false, aa, false, b1,
                                                     (short)0, acc1, false, false);
        acc2 = __builtin_amdgcn_wmma_f32_16x16x4_f32(false, at, false, b2,
                                                     (short)0, acc2, false, false);
        na = fmaf(aa.x, aa.x, fmaf(aa.y, aa.y, na));
        nt = fmaf(at.x, at.x, fmaf(at.y, at.y, nt));
    }

    // each lane pair (l, l^16) covers complementary K halves of row (l&15)
    na += __shfl_xor(na, 16, 32);
    nt += __shfl_xor(nt, 16, 32);

    // D col 0: lane 0 holds rows r0..r0+7 in acc[0..7], lane 16 holds r0+8..r0+15
    if (lane == 0) {
#pragma unroll
        for (int j = 0; j < 8; ++j) { sd1[wave][j] = acc1[j]; sd2[wave][j] = acc2[j]; }
    } else if (lane == 16) {
#pragma unroll
        for (int j = 0; j < 8; ++j) { sd1[wave][8 + j] = acc1[j]; sd2[wave][8 + j] = acc2[j]; }
    }
    __syncthreads();

    float u1 = 0.f, v1 = 0.f, u2 = 0.f, v2 = 0.f;
    if (lane < 16) {
        const int r = r0 + lane;
        const float q1 = sd1[wave][lane] / sqrtf(na);  // dot1 / ||s_a[r]||
        const float q2 = sd2[wave][lane] / sqrtf(nt);  // dot2 / ||s_t[r]||
        const int lab = label[r];
        const bool isPos = (lab == labA) && (r != anchor);
        const bool isNeg = (lab != labA);
        u1 = isPos ? q1 : 0.f;  v1 = isNeg ? q1 : 0.f;
        u2 = isPos ? q2 : 0.f;  v2 = isNeg ? q2 : 0.f;
    }
#pragma unroll
    for (int off = 16; off >= 1; off >>= 1) {
        u1 += __shfl_xor(u1, off, 32);
        v1 += __shfl_xor(v1, off, 32);
        u2 += __shfl_xor(u2, off, 32);
        v2 += __shfl_xor(v2, off, 32);
    }
    if (lane == 0) {
        sblk[wave][0] = u1; sblk[wave][1] = v1; sblk[wave][2] = u2; sblk[wave][3] = v2;
    }
    __syncthreads();

    if (threadIdx.x == 0) {   // fixed-order per-block combine -> deterministic
        float a0 = 0.f, a1 = 0.f, a2 = 0.f, a3 = 0.f;
#pragma unroll
        for (int w = 0; w < WAVES_PER_BLOCK; ++w) {
            a0 += sblk[w][0]; a1 += sblk[w][1]; a2 += sblk[w][2]; a3 += sblk[w][3];
        }
        float* p = partials + (size_t)blockIdx.x * 4;
        p[0] = a0; p[1] = a1; p[2] = a2; p[3] = a3;
    }
}

// ---------------- stage 5: anchor norms + final score ----------------
__global__ __launch_bounds__(256) void k_final(const float* __restrict__ s_t,
                                               const float* __restrict__ s_a,
                                               const int*   __restrict__ wsI,
                                               const float* __restrict__ partials,
                                               float*       __restrict__ out) {
    __shared__ float red[256];
    const int t = threadIdx.x;
    int anchor = wsI[8];
    if (anchor > BROWS - 1) anchor = 0;
    const float* at = s_t + (size_t)anchor * DDIM;
    const float* aa = s_a + (size_t)anchor * DDIM;

    float st = 0.f, sa = 0.f;
    for (int c = t; c < DDIM; c += 256) {
        float x = at[c]; st = fmaf(x, x, st);
        float y = aa[c]; sa = fmaf(y, y, sa);
    }
    red[t] = st; __syncthreads();
    for (int s = 128; s >= 1; s >>= 1) { if (t < s) red[t] += red[t + s]; __syncthreads(); }
    const float normT2 = red[0]; __syncthreads();
    red[t] = sa; __syncthreads();
    for (int s = 128; s >= 1; s >>= 1) { if (t < s) red[t] += red[t + s]; __syncthreads(); }
    const float normA2 = red[0];

    if (t == 0) {
        float u1 = 0.f, v1 = 0.f, u2 = 0.f, v2 = 0.f;
        for (int b = 0; b < MAIN_BLOCKS; ++b) {      // fixed order
            u1 += partials[b * 4 + 0]; v1 += partials[b * 4 + 1];
            u2 += partials[b * 4 + 2]; v2 += partials[b * 4 + 3];
        }
        const int   a       = wsI[7];
        const float pos_num = (float)wsI[a];         // includes anchor, as in reference
        const float neg_num = (float)BROWS - pos_num;
        const float nT = sqrtf(normT2);              // ||s_t[anchor]||
        const float nA = sqrtf(normA2);              // ||s_a[anchor]||
        const float pos1 = (u1 / nT) / pos_num;
        const float neg1 = (neg_num > 0.f) ? (v1 / nT) / fmaxf(neg_num, 1.f) : 0.f;
        const float pos2 = (u2 / nA) / pos_num;
        const float neg2 = (neg_num > 0.f) ? (v2 / nA) / fmaxf(neg_num, 1.f) : 0.f;
        const float sc1 = 2.f - pos1 + neg1;
        const float sc2 = 2.f - pos2 + neg2;
        out[0] = 0.5f * (sc1 + sc2);
    }
}

extern "C" void kernel_launch(void* const* d_in, const int* in_sizes, int n_in,
                              void* d_out, int out_size, void* d_ws, size_t ws_size,
                              hipStream_t stream) {
    const int*   label = (const int*)d_in[0];
    const float* s_t   = (const float*)d_in[1];
    const float* s_a   = (const float*)d_in[2];
    float* out = (float*)d_out;

    int*   wsI      = (int*)d_ws;          // [0..6]=counts, [7]=class, [8]=anchor
    float* partials = (float*)d_ws + 16;   // 64 blocks * 4 floats

    hipLaunchKernelGGL(k_init,   dim3(1),  dim3(64),  0, stream, wsI);
    hipLaunchKernelGGL(k_count,  dim3((BROWS + 255) / 256), dim3(256), 0, stream, label, wsI);
    hipLaunchKernelGGL(k_pick,   dim3(1),  dim3(1),   0, stream, wsI);
    hipLaunchKernelGGL(k_anchor, dim3((BROWS + 255) / 256), dim3(256), 0, stream, label, wsI);
    hipLaunchKernelGGL(k_main,   dim3(MAIN_BLOCKS), dim3(256), 0, stream,
                       s_t, s_a, label, wsI, partials);
    hipLaunchKernelGGL(k_final,  dim3(1),  dim3(256), 0, stream,
                       s_t, s_a, wsI, partials, out);
}